// SimpleModel_6725918785758
// MI455X (gfx1250) — compile-verified
//
#include <hip/hip_runtime.h>

typedef __attribute__((ext_vector_type(2))) float v2f;
typedef __attribute__((ext_vector_type(8))) float v8f;

// ---------------------------------------------------------------------------
// Sizes / layout
//   N = 4096 points, 256 tiles of 16 points.
//   Folded weight table Mw: 22528 logical rows x 16 cols (cols 7..15 zero),
//   stored PRE-SWIZZLED in WMMA B-operand lane order: for each K=4 chunk,
//   64 floats laid out so lane L reads its two B values (K = chunk*4 +
//   (L/16)*2 + {0,1}, N = L&15) as one contiguous b64 at chunkbase + L*2.
//     logical rows [    0, 4096)  : u(64) x v(64)  (c = a*64+b)
//     logical rows [ 4096, 5120)  : g(32) x h(32), runtime scale S3*|d|^2
//     logical rows [ 5120,10240)  : g x h, 5 blocks of 1024 (m), scale q_m(d)
//     logical rows [10240,16384)  : u(64) x h(32), 3 blocks of 2048, scale d_m
//     logical rows [16384,22528)  : g(32) x v(64), 3 blocks of 2048, scale d_m
//   Feature block per tile (stride 3328 floats, layout [ch][p], p=0..15):
//     ch   0.. 63 : u      ch  64..127 : v
//     ch 128..159 : g      ch 160..191 : h
//     ch 192..194 : d_m    ch 195..199 : q_m    ch 200 : S3*|d|^2
// ---------------------------------------------------------------------------
#define MW_ROWS      22528
#define FEAT_STRIDE  3328      // 208 * 16

__global__ void fold_weights(const float* __restrict__ W3_0,
                             const float* __restrict__ W3_1,
                             const float* __restrict__ W3_2,
                             const float* __restrict__ Wout,
                             float* __restrict__ Mw) {
    int idx = blockIdx.x * blockDim.x + threadIdx.x;
    if (idx >= MW_ROWS * 16) return;
    int row = idx >> 4;
    int o   = idx & 15;
    float acc = 0.0f;
    if (o < 7) {
        if (row < 5120) {                       // uv (c<4096) and gh0 (4096..5119)
            for (int dd = 0; dd < 64; ++dd)
                acc += W3_0[row * 64 + dd] * Wout[dd * 7 + o];
        } else if (row < 10240) {               // g x h with q_m weights (y2 path)
            int m = (row - 5120) >> 10;
            int i = (row - 5120) & 1023;
            for (int dd = 0; dd < 16; ++dd)
                acc += W3_2[i * 16 + dd] * Wout[(160 + dd * 5 + m) * 7 + o];
        } else if (row < 16384) {               // u x h with d_m weights (y1, p011)
            int m = (row - 10240) >> 11;
            int i = (row - 10240) & 2047;
            for (int dd = 0; dd < 32; ++dd)
                acc += W3_1[i * 32 + dd] * Wout[(64 + dd * 3 + m) * 7 + o];
        } else {                                // g x v with d_m weights (y1, p101)
            int m = (row - 16384) >> 11;
            int i = (row - 16384) & 2047;
            for (int dd = 0; dd < 32; ++dd)
                acc += W3_1[(2048 + i) * 32 + dd] * Wout[(64 + dd * 3 + m) * 7 + o];
        }
    }
    // Swizzle into B-operand lane order: chunk*64 + ((row>>1 & 1)*16 + o)*2 + (row&1)
    int sw = ((row >> 2) << 6) + ((((row >> 1) & 1) * 16 + o) << 1) + (row & 1);
    Mw[sw] = acc;
}

// ---------------------------------------------------------------------------
// prep: per tile of 16 points, compute RBFs and the 4 channel-mixed feature
// vectors u,v,g,h plus d_m, q_m(d), S3*|d|^2, stored [ch][p].
// ---------------------------------------------------------------------------
__global__ void prep(const float* __restrict__ pos,
                     const float* __restrict__ W1_0,
                     const float* __restrict__ W1_1,
                     const float* __restrict__ W2_0,
                     const float* __restrict__ W2_1,
                     float* __restrict__ feat) {
    __shared__ float flds[96 * 16];   // f0 rows 0..63, f1 rows 64..95, [c][p]
    __shared__ float rlds[16];
    const int tile = blockIdx.x;
    const int t    = threadIdx.x;
    float* fb = feat + (size_t)tile * FEAT_STRIDE;

    if (t < 16) {
        int n = tile * 16 + t;
        float x = pos[n * 3 + 0], y = pos[n * 3 + 1], z = pos[n * 3 + 2];
        float r = sqrtf(x * x + y * y + z * z) + 1e-9f;
        rlds[t] = r;
        float d0 = y / r, d1 = z / r, d2 = x / r;        // (y,z,x) order
        const float S2v = 0.70710678118654752f;          // 1/sqrt(2)
        const float S3v = 0.57735026918962576f;          // 1/sqrt(3)
        const float S6v = 0.40824829046386302f;          // 1/sqrt(6)
        fb[(192 + 0) * 16 + t] = d0;
        fb[(192 + 1) * 16 + t] = d1;
        fb[(192 + 2) * 16 + t] = d2;
        fb[(195 + 0) * 16 + t] = 2.0f * S2v * d0 * d2;                      // ~xy
        fb[(195 + 1) * 16 + t] = 2.0f * S2v * d0 * d1;                      // ~yz
        fb[(195 + 2) * 16 + t] = S6v * (2.0f * d1 * d1 - d2 * d2 - d0 * d0);// ~2z^2-..
        fb[(195 + 3) * 16 + t] = 2.0f * S2v * d2 * d1;                      // ~xz
        fb[(195 + 4) * 16 + t] = S2v * (d2 * d2 - d0 * d0);                 // ~x^2-y^2
        fb[200 * 16 + t] = S3v * (d0 * d0 + d1 * d1 + d2 * d2);             // p110 scale
    }
    __syncthreads();

    // Gaussian RBFs: 96 channels x 16 points
    for (int e = t; e < 96 * 16; e += 256) {
        int p = e & 15, c = e >> 4;
        float r  = rlds[p];
        float cc = (c < 64) ? ((float)c * (5.0f / 63.0f))
                            : ((float)(c - 64) * (5.0f / 31.0f));
        float dr = r - cc;
        flds[e] = __expf(-4.0f * dr * dr);
    }
    __syncthreads();

    // Channel mixing: 192 output channels x 16 points = 3072 dots
    for (int e = t; e < 192 * 16; e += 256) {
        int p = e & 15, ch = e >> 4;
        float acc = 0.0f;
        if (ch < 64) {
            for (int c = 0; c < 64; ++c) acc += flds[c * 16 + p] * W1_0[c * 64 + ch];
        } else if (ch < 128) {
            int d = ch - 64;
            for (int c = 0; c < 64; ++c) acc += flds[c * 16 + p] * W2_0[c * 64 + d];
        } else if (ch < 160) {
            int d = ch - 128;
            for (int c = 0; c < 32; ++c) acc += flds[(64 + c) * 16 + p] * W1_1[c * 32 + d];
        } else {
            int d = ch - 160;
            for (int c = 0; c < 32; ++c) acc += flds[(64 + c) * 16 + p] * W2_1[c * 32 + d];
        }
        fb[ch * 16 + p] = acc;
    }
}

// ---------------------------------------------------------------------------
// tp_gemm: fused on-the-fly tensor product + folded-weight GEMM via
// V_WMMA_F32_16X16X4_F32. One block = 16 points, 8 waves split K; each wave
// keeps two alternating 16x16 f32 accumulators covering all 13 bilinear
// families (row scales folded into the A operand). The a-outer/b-inner loop
// hoists the X value; B comes as a single coalesced b64 per chunk from the
// pre-swizzled weight table. Deterministic LDS reduction at the end.
// ---------------------------------------------------------------------------
__global__ void __launch_bounds__(256)
tp_gemm(const float* __restrict__ feat,
        const float* __restrict__ Mw,
        const float* __restrict__ b_out,
        float* __restrict__ out) {
    __shared__ float F[FEAT_STRIDE];   // 13.3 KB features [ch][p]
    __shared__ float red[8 * 256];     // per-wave partial tiles

    const int tile = blockIdx.x;
    const int t    = threadIdx.x;
    const float* fb = feat + (size_t)tile * FEAT_STRIDE;
    for (int e = t; e < FEAT_STRIDE; e += 256) F[e] = fb[e];
    __syncthreads();

    const int lane = t & 31;
    const int w    = t >> 5;
    const int p    = lane & 15;            // point (A row / M)
    const int koff = (lane >> 4) << 1;     // lanes 0-15 hold K0/K1, 16-31 hold K2/K3

    v8f acc0 = {0.f, 0.f, 0.f, 0.f, 0.f, 0.f, 0.f, 0.f};
    v8f acc1 = {0.f, 0.f, 0.f, 0.f, 0.f, 0.f, 0.f, 0.f};

    // families: 0=uv  1=gh0  2..6=gh*q_m  7..9=uh*d_m  10..12=gv*d_m
    for (int f = 0; f < 13; ++f) {
        int Xo, Yo, rowbase, acount, bchunks;
        float s;
        if (f == 0)      { Xo = 0;    Yo = 1024; rowbase = 0;                    acount = 8; bchunks = 16; s = 1.0f; }
        else if (f == 1) { Xo = 2048; Yo = 2560; rowbase = 4096;                 acount = 4; bchunks = 8;  s = F[200 * 16 + p]; }
        else if (f < 7)  { int m = f - 2;  Xo = 2048; Yo = 2560; rowbase = 5120  + m * 1024; acount = 4; bchunks = 8;  s = F[(195 + m) * 16 + p]; }
        else if (f < 10) { int m = f - 7;  Xo = 0;    Yo = 2560; rowbase = 10240 + m * 2048; acount = 8; bchunks = 8;  s = F[(192 + m) * 16 + p]; }
        else             { int m = f - 10; Xo = 2048; Yo = 1024; rowbase = 16384 + m * 2048; acount = 4; bchunks = 16; s = F[(192 + m) * 16 + p]; }

        const int astart = w * acount;
        const float* X   = &F[Xo + astart * 16 + p];
        const float* Yb  = &F[Yo + koff * 16 + p];       // +bc*64 walks b in steps of 4
        // swizzled weights: 64 floats per K=4 chunk; this wave's slice starts at
        // chunk (rowbase/4 + w*acount*bchunks); lane reads b64 at +lane*2.
        const float* WB  = Mw + ((size_t)(rowbase >> 2) + (size_t)w * acount * bchunks) * 64
                              + (size_t)lane * 2;

        for (int ai = 0; ai < acount; ++ai) {
            const float xv = X[ai * 16] * s;
            const float* WBi = WB + (size_t)ai * bchunks * 64;
            #pragma unroll 4
            for (int bc = 0; bc < bchunks; ++bc) {
                v2f A;
                A.x = xv * Yb[bc * 64];          // Y[(bc*4+koff)  ][p]
                A.y = xv * Yb[bc * 64 + 16];     // Y[(bc*4+koff)+1][p]
                v2f B = *(const v2f*)(WBi + (size_t)bc * 64);
                if ((bc & 1) == 0)
                    acc0 = __builtin_amdgcn_wmma_f32_16x16x4_f32(
                        false, A, false, B, (short)0, acc0, false, false);
                else
                    acc1 = __builtin_amdgcn_wmma_f32_16x16x4_f32(
                        false, A, false, B, (short)0, acc1, false, false);
            }
        }
    }

    // Deterministic cross-wave reduction: stage each wave's tile, sum in order.
    const v8f accs = acc0 + acc1;
    const int mbase = (lane >> 4) * 8;       // D layout: VGPR j -> M = j + 8*(lane/16)
    float* rw = &red[w * 256];
    #pragma unroll
    for (int j = 0; j < 8; ++j) rw[(mbase + j) * 16 + p] = accs[j];
    __syncthreads();

    const int M = t >> 4;
    const int Ncol = t & 15;
    float sum = 0.0f;
    #pragma unroll
    for (int ww = 0; ww < 8; ++ww) sum += red[ww * 256 + t];
    if (Ncol < 7)
        out[((size_t)tile * 16 + M) * 7 + Ncol] = sum + b_out[Ncol];
}

// ---------------------------------------------------------------------------
extern "C" void kernel_launch(void* const* d_in, const int* in_sizes, int n_in,
                              void* d_out, int out_size, void* d_ws, size_t ws_size,
                              hipStream_t stream) {
    const float* pos  = (const float*)d_in[0];
    const float* W1_0 = (const float*)d_in[1];
    const float* W1_1 = (const float*)d_in[2];
    const float* W2_0 = (const float*)d_in[3];
    const float* W2_1 = (const float*)d_in[4];
    const float* W3_0 = (const float*)d_in[5];
    const float* W3_1 = (const float*)d_in[6];
    const float* W3_2 = (const float*)d_in[7];
    const float* Wout = (const float*)d_in[8];
    const float* bout = (const float*)d_in[9];

    float* Mw   = (float*)d_ws;                 // 22528*16 floats = 1.44 MB
    float* feat = Mw + (size_t)MW_ROWS * 16;    // 256*3328 floats = 3.41 MB

    fold_weights<<<(MW_ROWS * 16 + 255) / 256, 256, 0, stream>>>(W3_0, W3_1, W3_2, Wout, Mw);
    prep<<<256, 256, 0, stream>>>(pos, W1_0, W1_1, W2_0, W2_1, feat);
    tp_gemm<<<256, 256, 0, stream>>>(feat, Mw, bout, (float*)d_out);
}